// HGTRec_59725815218510
// MI455X (gfx1250) — compile-verified
//
#include <hip/hip_runtime.h>

typedef float v2f __attribute__((ext_vector_type(2)));
typedef float v8f __attribute__((ext_vector_type(8)));

#define DD 128
#define HH 8

// ---------------------------------------------------------------------------
// fp32 WMMA: D(16x16) += A(16x4) * B(4x16).  ISA layouts (CDNA5 ISA):
//   A: lane m = lane&15 holds row M=m; VGPR j, lane-half g -> K = 2g + j
//   B: lane n = lane&15 holds col N=n; VGPR j, lane-half g -> K = 2g + j
//   C/D: lane n holds col N=n; VGPR r -> row M = r + 8g
// ---------------------------------------------------------------------------
__device__ __forceinline__ v8f wmma4(v2f a, v2f b, v8f c) {
  return __builtin_amdgcn_wmma_f32_16x16x4_f32(false, a, false, b, (short)0, c,
                                               false, false);
}

// C[N,128] = act(A[N,128] @ W[128,128] + bias[128]); act: 0=none, 1=exact GELU
__global__ __launch_bounds__(256) void gemm128_kernel(
    const float* __restrict__ A, const float* __restrict__ W,
    const float* __restrict__ bias, float* __restrict__ C, int N, int act) {
  __shared__ float Alds[16 * 132];  // +4 pad: conflict-free fragment reads
  const int tid = threadIdx.x;
  const int row0 = blockIdx.x * 16;
  // stage 16x128 A tile (float4, coalesced)
  for (int i = tid; i < 512; i += 256) {
    int r = i >> 5;
    int c4 = (i & 31) << 2;
    int row = row0 + r;
    float4 v = make_float4(0.f, 0.f, 0.f, 0.f);
    if (row < N) v = *(const float4*)(A + (size_t)row * DD + c4);
    float* dst = &Alds[r * 132 + c4];
    dst[0] = v.x; dst[1] = v.y; dst[2] = v.z; dst[3] = v.w;
  }
  __syncthreads();
  const int wave = tid >> 5;
  const int lane = tid & 31;
  const int col0 = wave * 16;     // each wave: 16 output columns
  const int g = lane >> 4;
  const int n = lane & 15;        // A-row m and B/C-col n coincide per layout
  v8f acc0 = {0, 0, 0, 0, 0, 0, 0, 0};
  v8f acc1 = {0, 0, 0, 0, 0, 0, 0, 0};
#pragma unroll
  for (int k0 = 0; k0 < DD; k0 += 8) {
    {
      v2f a, b;
      const float* ap = &Alds[n * 132 + k0 + 2 * g];
      a.x = ap[0]; a.y = ap[1];
      const float* bp = W + (size_t)(k0 + 2 * g) * DD + col0 + n;
      b.x = bp[0]; b.y = bp[DD];
      acc0 = wmma4(a, b, acc0);
    }
    {
      v2f a, b;
      const float* ap = &Alds[n * 132 + k0 + 4 + 2 * g];
      a.x = ap[0]; a.y = ap[1];
      const float* bp = W + (size_t)(k0 + 4 + 2 * g) * DD + col0 + n;
      b.x = bp[0]; b.y = bp[DD];
      acc1 = wmma4(a, b, acc1);
    }
  }
  const float bv = bias[col0 + n];
#pragma unroll
  for (int r = 0; r < 8; ++r) {
    int row = row0 + r + 8 * g;
    if (row < N) {
      float x = acc0[r] + acc1[r] + bv;
      if (act == 1) x = 0.5f * x * (1.0f + erff(x * 0.70710678118654752f));
      C[(size_t)row * DD + col0 + n] = x;
    }
  }
}

// W_eff[d, h*16+e] = sum_dd Wsrc[d, h*16+dd] * rel[h,dd,e]; same for bias
__global__ __launch_bounds__(256) void build_weff_kernel(
    const float* __restrict__ Wsrc, const float* __restrict__ bsrc,
    const float* __restrict__ rel, float* __restrict__ Weff,
    float* __restrict__ beff) {
  int idx = blockIdx.x * 256 + threadIdx.x;
  if (idx < DD * DD) {
    int d = idx >> 7, c = idx & 127;
    int h = c >> 4, e = c & 15;
    const float* rp = rel + h * 256 + e;          // rel[h][dd][e], stride 16
    const float* wp = Wsrc + (size_t)d * DD + h * 16;
    float s = 0.f;
#pragma unroll
    for (int t = 0; t < 16; ++t) s += wp[t] * rp[t * 16];
    Weff[idx] = s;
  } else if (idx < DD * DD + DD) {
    int c = idx - DD * DD;
    int h = c >> 4, e = c & 15;
    const float* rp = rel + h * 256 + e;
    const float* bp = bsrc + h * 16;
    float s = 0.f;
#pragma unroll
    for (int t = 0; t < 16; ++t) s += bp[t] * rp[t * 16];
    beff[c] = s;
  }
}

// monotone float <-> uint mapping for atomic segment-max
__device__ __forceinline__ unsigned fmap(float f) {
  unsigned b = __float_as_uint(f);
  return (b & 0x80000000u) ? ~b : (b | 0x80000000u);
}
__device__ __forceinline__ float funmap(unsigned u) {
  return __uint_as_float((u & 0x80000000u) ? (u ^ 0x80000000u) : ~u);
}

// wave per edge: s[e,h] = <Q[dst],K[src]>_h * pri[h]/4; atomicMax seg-max
__global__ __launch_bounds__(256) void edge_scores_kernel(
    const float* __restrict__ Q, const float* __restrict__ K,
    const int* __restrict__ esrc, const int* __restrict__ edst,
    const float* __restrict__ pri, float* __restrict__ sout,
    unsigned* __restrict__ mmap, int nE) {
  int wid = (blockIdx.x * 256 + threadIdx.x) >> 5;
  int lane = threadIdx.x & 31;
  if (wid >= nE) return;
  int sn = esrc[wid], dn = edst[wid];
  float4 q = *(const float4*)(Q + (size_t)dn * DD + lane * 4);
  float4 k = *(const float4*)(K + (size_t)sn * DD + lane * 4);
  float p = q.x * k.x + q.y * k.y + q.z * k.z + q.w * k.w;
  p += __shfl_xor(p, 1, 32);
  p += __shfl_xor(p, 2, 32);   // now per-head sum in each 4-lane group
  int h = lane >> 2;
  float score = p * pri[h] * 0.25f;  // 1/sqrt(DK)=1/4
  if ((lane & 3) == 0) {
    sout[(size_t)wid * HH + h] = score;
    atomicMax(mmap + (size_t)dn * HH + h, fmap(score));
  }
}

// wave per edge: ex = exp(s-m); numer[dst] += ex*V[src]; den[dst,h] += ex
__global__ __launch_bounds__(256) void edge_accum_kernel(
    const float* __restrict__ sin, const unsigned* __restrict__ mmap,
    const float* __restrict__ V, const int* __restrict__ esrc,
    const int* __restrict__ edst, float* __restrict__ numer,
    float* __restrict__ den, int nE) {
  int wid = (blockIdx.x * 256 + threadIdx.x) >> 5;
  int lane = threadIdx.x & 31;
  if (wid >= nE) return;
  int sn = esrc[wid], dn = edst[wid];
  int h = lane >> 2;
  float sc = sin[(size_t)wid * HH + h];
  float m = funmap(mmap[(size_t)dn * HH + h]);
  float ex = expf(sc - m);
  float4 v = *(const float4*)(V + (size_t)sn * DD + lane * 4);
  float* np = numer + (size_t)dn * DD + lane * 4;
  atomicAdd(np + 0, ex * v.x);
  atomicAdd(np + 1, ex * v.y);
  atomicAdd(np + 2, ex * v.z);
  atomicAdd(np + 3, ex * v.w);
  if ((lane & 3) == 0) atomicAdd(den + (size_t)dn * HH + h, ex);
}

__global__ __launch_bounds__(256) void finalize_kernel(
    float* __restrict__ t, const float* __restrict__ den, int count) {
  int i = blockIdx.x * 256 + threadIdx.x;
  if (i >= count) return;
  int n = i >> 7, c = i & 127;
  float d = den[n * HH + (c >> 4)];
  t[i] = (d > 0.f) ? t[i] / d : 0.f;  // isolated node -> 0 (matches segsum)
}

// wave per row: x = tr*sig(skip) + h*(1-sig); LayerNorm(x)*g + b
__global__ __launch_bounds__(256) void ln_out_kernel(
    const float* __restrict__ tr, const float* __restrict__ hprev,
    const float* __restrict__ skipv, const float* __restrict__ gam,
    const float* __restrict__ bet, float* __restrict__ outp, int N) {
  int wid = (blockIdx.x * 256 + threadIdx.x) >> 5;
  int lane = threadIdx.x & 31;
  if (wid >= N) return;
  float alpha = 1.f / (1.f + expf(-skipv[0]));
  float beta = 1.f - alpha;
  size_t base = (size_t)wid * DD + lane * 4;
  float4 t = *(const float4*)(tr + base);
  float4 h = *(const float4*)(hprev + base);
  float x0 = t.x * alpha + h.x * beta;
  float x1 = t.y * alpha + h.y * beta;
  float x2 = t.z * alpha + h.z * beta;
  float x3 = t.w * alpha + h.w * beta;
  float s = x0 + x1 + x2 + x3;
#pragma unroll
  for (int off = 1; off < 32; off <<= 1) s += __shfl_xor(s, off, 32);
  float mu = s * (1.f / 128.f);
  float d0 = x0 - mu, d1 = x1 - mu, d2 = x2 - mu, d3 = x3 - mu;
  float v = d0 * d0 + d1 * d1 + d2 * d2 + d3 * d3;
#pragma unroll
  for (int off = 1; off < 32; off <<= 1) v += __shfl_xor(v, off, 32);
  float inv = rsqrtf(v * (1.f / 128.f) + 1e-5f);
  float4 gg = *(const float4*)(gam + lane * 4);
  float4 bb = *(const float4*)(bet + lane * 4);
  float4 o;
  o.x = d0 * inv * gg.x + bb.x;
  o.y = d1 * inv * gg.y + bb.y;
  o.z = d2 * inv * gg.z + bb.z;
  o.w = d3 * inv * gg.w + bb.w;
  *(float4*)(outp + base) = o;
}

// wave per (b,p): gather user/pos/neg rows, dots + emit 5 outputs
__global__ __launch_bounds__(256) void score_kernel(
    const float* __restrict__ hu, const float* __restrict__ hi,
    const int* __restrict__ uid, const int* __restrict__ pos,
    const int* __restrict__ neg, float* __restrict__ pos_logits,
    float* __restrict__ neg_logits, float* __restrict__ ue2,
    float* __restrict__ pos_emb, float* __restrict__ neg_emb, int BP, int P,
    int NN) {
  int wid = (blockIdx.x * 256 + threadIdx.x) >> 5;
  int lane = threadIdx.x & 31;
  if (wid >= BP) return;
  int b = wid / P;
  float4 u = *(const float4*)(hu + (size_t)uid[b] * DD + lane * 4);
  float4 pe = *(const float4*)(hi + (size_t)pos[wid] * DD + lane * 4);
  *(float4*)(pos_emb + (size_t)wid * DD + lane * 4) = pe;
  float dp = u.x * pe.x + u.y * pe.y + u.z * pe.z + u.w * pe.w;
#pragma unroll
  for (int off = 1; off < 32; off <<= 1) dp += __shfl_xor(dp, off, 32);
  if (lane == 0) pos_logits[wid] = dp;
  const int* ng = neg + (size_t)wid * NN;
  for (int j = 0; j < NN; ++j) {
    float4 ne = *(const float4*)(hi + (size_t)ng[j] * DD + lane * 4);
    size_t base = ((size_t)wid * NN + j) * DD + lane * 4;
    *(float4*)(neg_emb + base) = ne;
    *(float4*)(ue2 + base) = u;
    float dn = u.x * ne.x + u.y * ne.y + u.z * ne.z + u.w * ne.w;
#pragma unroll
    for (int off = 1; off < 32; off <<= 1) dn += __shfl_xor(dn, off, 32);
    if (lane == 0) neg_logits[(size_t)wid * NN + j] = dn;
  }
}

extern "C" void kernel_launch(void* const* d_in, const int* in_sizes, int n_in,
                              void* d_out, int out_size, void* d_ws,
                              size_t ws_size, hipStream_t stream) {
  const float* emb_user = (const float*)d_in[0];
  const float* emb_item = (const float*)d_in[1];
  const float* W_adapt = (const float*)d_in[2];
  const float* b_adapt = (const float*)d_in[3];
  const float* Wk = (const float*)d_in[4];
  const float* bk = (const float*)d_in[5];
  const float* Wq = (const float*)d_in[6];
  const float* bq = (const float*)d_in[7];
  const float* Wv = (const float*)d_in[8];
  const float* bv = (const float*)d_in[9];
  const float* Wa = (const float*)d_in[10];
  const float* ba = (const float*)d_in[11];
  const float* ln_g = (const float*)d_in[12];
  const float* ln_b = (const float*)d_in[13];
  const float* rel_pri = (const float*)d_in[14];
  const float* rel_att = (const float*)d_in[15];
  const float* rel_msg = (const float*)d_in[16];
  const float* skip = (const float*)d_in[17];
  const int* e_u = (const int*)d_in[18];
  const int* e_i = (const int*)d_in[19];
  const int* uid = (const int*)d_in[20];
  const int* pos = (const int*)d_in[21];
  const int* neg = (const int*)d_in[22];
  float* out = (float*)d_out;

  const int NU = in_sizes[0] / DD;
  const int NI = in_sizes[1] / DD;
  const int Ecnt = in_sizes[18];
  const int Bb = in_sizes[20];
  const int Pp = in_sizes[21] / Bb;
  const int NN = in_sizes[22] / (Bb * Pp);
  const int BP = Bb * Pp;

  // ---- workspace carve-out (floats) ----
  float* w = (float*)d_ws;
  size_t o = 0;
  auto alloc = [&](size_t nf) { float* p = w + o; o += nf; return p; };
  float* hA_u = alloc((size_t)NU * DD);
  float* hA_i = alloc((size_t)NI * DD);
  float* hB_u = alloc((size_t)NU * DD);
  float* hB_i = alloc((size_t)NI * DD);
  float* Ku = alloc((size_t)NU * DD);
  float* Vu = alloc((size_t)NU * DD);
  float* Qu = alloc((size_t)NU * DD);
  float* Ki = alloc((size_t)NI * DD);
  float* Vi = alloc((size_t)NI * DD);
  float* Qi = alloc((size_t)NI * DD);
  float* sbuf = alloc((size_t)Ecnt * HH);
  unsigned* m_u = (unsigned*)alloc((size_t)NU * HH);
  float* den_u = alloc((size_t)NU * HH);
  unsigned* m_i = (unsigned*)alloc((size_t)NI * HH);
  float* den_i = alloc((size_t)NI * HH);
  float* wke0 = alloc(DD * DD); float* bke0 = alloc(DD);
  float* wve0 = alloc(DD * DD); float* bve0 = alloc(DD);
  float* wke1 = alloc(DD * DD); float* bke1 = alloc(DD);
  float* wve1 = alloc(DD * DD); float* bve1 = alloc(DD);
  float* tmp_u = Ku;  // K/V/Q are dead by output-transform time
  float* tmp_i = Ki;

  const int gu = (NU + 15) / 16, gi = (NI + 15) / 16;
  const int eb = (Ecnt + 7) / 8;           // 8 edge-waves per 256-thr block
  const int wu = (NU + 7) / 8, wi = (NI + 7) / 8;  // wave-per-row grids

  // ---- adapt: h = gelu(emb @ W_adapt + b) ----
  gemm128_kernel<<<gu, 256, 0, stream>>>(emb_user, W_adapt, b_adapt, hA_u, NU, 1);
  gemm128_kernel<<<gi, 256, 0, stream>>>(emb_item, W_adapt + 16384, b_adapt + 128,
                                         hA_i, NI, 1);

  float* cu = hA_u; float* ci = hA_i;   // current
  float* ou = hB_u; float* oi = hB_i;   // next (also numer accumulator)

  for (int l = 0; l < 2; ++l) {
    const size_t L0 = (size_t)(l * 2 + 0), L1 = (size_t)(l * 2 + 1);
    // effective K/V weights: W @ blockdiag(rel)
    build_weff_kernel<<<65, 256, 0, stream>>>(Wk + L0 * 16384, bk + L0 * 128,
                                              rel_att + L0 * 2048, wke0, bke0);
    build_weff_kernel<<<65, 256, 0, stream>>>(Wv + L0 * 16384, bv + L0 * 128,
                                              rel_msg + L0 * 2048, wve0, bve0);
    build_weff_kernel<<<65, 256, 0, stream>>>(Wk + L1 * 16384, bk + L1 * 128,
                                              rel_att + L1 * 2048, wke1, bke1);
    build_weff_kernel<<<65, 256, 0, stream>>>(Wv + L1 * 16384, bv + L1 * 128,
                                              rel_msg + L1 * 2048, wve1, bve1);
    // projections (WMMA GEMMs)
    gemm128_kernel<<<gu, 256, 0, stream>>>(cu, wke0, bke0, Ku, NU, 0);
    gemm128_kernel<<<gu, 256, 0, stream>>>(cu, wve0, bve0, Vu, NU, 0);
    gemm128_kernel<<<gu, 256, 0, stream>>>(cu, Wq + L0 * 16384, bq + L0 * 128, Qu, NU, 0);
    gemm128_kernel<<<gi, 256, 0, stream>>>(ci, wke1, bke1, Ki, NI, 0);
    gemm128_kernel<<<gi, 256, 0, stream>>>(ci, wve1, bve1, Vi, NI, 0);
    gemm128_kernel<<<gi, 256, 0, stream>>>(ci, Wq + L1 * 16384, bq + L1 * 128, Qi, NI, 0);

    // relation 0: users -> items (dst = items)
    hipMemsetAsync(m_i, 0, (size_t)NI * HH * 4, stream);
    hipMemsetAsync(den_i, 0, (size_t)NI * HH * 4, stream);
    hipMemsetAsync(oi, 0, (size_t)NI * DD * 4, stream);
    edge_scores_kernel<<<eb, 256, 0, stream>>>(Qi, Ku, e_u, e_i,
                                               rel_pri + L0 * 8, sbuf, m_i, Ecnt);
    edge_accum_kernel<<<eb, 256, 0, stream>>>(sbuf, m_i, Vu, e_u, e_i, oi, den_i, Ecnt);
    finalize_kernel<<<(NI * DD + 255) / 256, 256, 0, stream>>>(oi, den_i, NI * DD);

    // relation 1: items -> users (dst = users)
    hipMemsetAsync(m_u, 0, (size_t)NU * HH * 4, stream);
    hipMemsetAsync(den_u, 0, (size_t)NU * HH * 4, stream);
    hipMemsetAsync(ou, 0, (size_t)NU * DD * 4, stream);
    edge_scores_kernel<<<eb, 256, 0, stream>>>(Qu, Ki, e_i, e_u,
                                               rel_pri + L1 * 8, sbuf, m_u, Ecnt);
    edge_accum_kernel<<<eb, 256, 0, stream>>>(sbuf, m_u, Vi, e_i, e_u, ou, den_u, Ecnt);
    finalize_kernel<<<(NU * DD + 255) / 256, 256, 0, stream>>>(ou, den_u, NU * DD);

    // output transform + skip + LayerNorm (LN overwrites numer in place)
    gemm128_kernel<<<gu, 256, 0, stream>>>(ou, Wa + L0 * 16384, ba + L0 * 128, tmp_u, NU, 0);
    ln_out_kernel<<<wu, 256, 0, stream>>>(tmp_u, cu, skip + L0, ln_g + L0 * 128,
                                          ln_b + L0 * 128, ou, NU);
    gemm128_kernel<<<gi, 256, 0, stream>>>(oi, Wa + L1 * 16384, ba + L1 * 128, tmp_i, NI, 0);
    ln_out_kernel<<<wi, 256, 0, stream>>>(tmp_i, ci, skip + L1, ln_g + L1 * 128,
                                          ln_b + L1 * 128, oi, NI);
    // ping-pong
    float* t;
    t = cu; cu = ou; ou = t;
    t = ci; ci = oi; oi = t;
  }

  // ---- final gather + logits; outputs concatenated flat ----
  float* pos_logits = out;                                   // [B,P,1]
  float* neg_logits = pos_logits + (size_t)BP;               // [B,P,NN]
  float* ue2 = neg_logits + (size_t)BP * NN;                 // [B,P,NN,D]
  float* pos_emb = ue2 + (size_t)BP * NN * DD;               // [B,P,D]
  float* neg_emb = pos_emb + (size_t)BP * DD;                // [B,P,NN,D]
  score_kernel<<<(BP + 7) / 8, 256, 0, stream>>>(cu, ci, uid, pos, neg,
                                                 pos_logits, neg_logits, ue2,
                                                 pos_emb, neg_emb, BP, Pp, NN);
}